// cpu_multi_cal_grad_47090021433587
// MI455X (gfx1250) — compile-verified
//
#include <hip/hip_runtime.h>
#include <hip/hip_bf16.h>
#include <math.h>
#include <stdint.h>

#define MAXL 64
#define BLK  128
#define TWO_PI_F 6.2831853071795864f
#define LN01_F  -2.3025850929940457f  // ln(0.1)

__device__ __forceinline__ float rcpf(float x)  { return __builtin_amdgcn_rcpf(x); }
__device__ __forceinline__ float sqrt_f(float x){ return __builtin_amdgcn_sqrtf(x); }

// ---- last-layer initialization of the e-vector -----------------------------
__device__ __forceinline__ void dltar_init(float wvno2, float xka, float xkb,
                                           float gammk, float r, float e[5])
{
    float ra = sqrt_f(fabsf(wvno2 - xka * xka));
    float rb = sqrt_f(fabsf(wvno2 - xkb * xkb));
    float gam   = gammk * wvno2;
    float gamm1 = gam - 1.f;
    e[0] = r * r * (gamm1 * gamm1 - gam * gammk * ra * rb);
    e[1] = -r * ra;
    e[2] = r * (gamm1 - gammk * ra * rb);
    e[3] = r * rb;
    e[4] = wvno2 - ra * rb;
}

// ---- one layer step: _var + _dnka + e <- e*CA (+ renorm) -------------------
__device__ __forceinline__ void dltar_step(float wvno, float wvno2,
    float xka, float xkb, float gammk, float rho, float dm, float e[5])
{
    float gam = gammk * wvno2;
    float ra  = sqrt_f(fabsf(wvno2 - xka * xka));
    float rb  = sqrt_f(fabsf(wvno2 - xkb * xkb));
    float p   = ra * dm, q = rb * dm;

    // ---------- _var ----------
    float ra_s   = (ra > 0.f) ? ra : 1.f;
    float rb_s   = (rb > 0.f) ? rb : 1.f;
    float sinp_o = __sinf(p);
    float fac_p  = (p < 16.f) ? __expf(-2.f * p) : 0.f;
    float cosp_e = (1.f + fac_p) * 0.5f;
    float sinp_e = (1.f - fac_p) * 0.5f;
    bool  lt_a = wvno < xka, gt_a = wvno > xka;
    float pex  = gt_a ? p : 0.f;
    float cosp = lt_a ? __cosf(p) : (gt_a ? cosp_e : 1.f);
    float inv_ra = rcpf(ra_s);
    float w = lt_a ? sinp_o * inv_ra : (gt_a ? sinp_e * inv_ra : dm);
    float x = lt_a ? (-ra * sinp_o)  : (gt_a ? ra * sinp_e    : 0.f);

    float sinq_o = __sinf(q);
    float fac_q  = (q < 16.f) ? __expf(-2.f * q) : 0.f;
    float cosq_e = (1.f + fac_q) * 0.5f;
    float sinq_e = (1.f - fac_q) * 0.5f;
    bool  lt_b = wvno < xkb, gt_b = wvno > xkb;
    float sex  = gt_b ? q : 0.f;
    float cosq = lt_b ? __cosf(q) : (gt_b ? cosq_e : 1.f);
    float inv_rb = rcpf(rb_s);
    float y = lt_b ? sinq_o * inv_rb : (gt_b ? sinq_e * inv_rb : dm);
    float z = lt_b ? (-rb * sinq_o)  : (gt_b ? rb * sinq_e    : 0.f);

    float exa = pex + sex;
    float a0  = (exa < 60.f) ? __expf(-exa) : 0.f;

    float cpcq = cosp * cosq, cpy = cosp * y, cpz = cosp * z;
    float cqw  = cosq * w,    cqx = cosq * x;
    float xy = x * y, xz = x * z, wy = w * y, wz = w * z;

    // ---------- _dnka ----------
    float gamm1 = gam - 1.f;
    float twgm1 = gam + gamm1;
    float gmgmk = gam * gammk;
    float gmgm1 = gam * gamm1;
    float gm1sq = gamm1 * gamm1;
    float invr  = rcpf(rho);
    float rho2  = rho * rho;
    float invr2 = invr * invr;
    float a0pq  = a0 - cpcq;
    float t     = -2.f * wvno2;

    float c00 = cpcq - 2.f * gmgm1 * a0pq - gmgmk * xz - wvno2 * gm1sq * wy;
    float c01 = (wvno2 * cpy - cqx) * invr;
    float c02 = -(twgm1 * a0pq + gammk * xz + wvno2 * gamm1 * wy) * invr;
    float c03 = (cpz - wvno2 * cqw) * invr;
    float c04 = -(2.f * wvno2 * a0pq + xz + wvno2 * wvno2 * wy) * invr2;
    float c10 = (gmgmk * cpz - gm1sq * cqw) * rho;
    float c11 = cpcq;
    float c12 = gammk * cpz - gamm1 * cqw;
    float c13 = -wz;
    float c30 = (gm1sq * cpy - gmgmk * cqx) * rho;
    float c31 = -xy;
    float c32 = gamm1 * cpy - gammk * cqx;
    float c40 = -(2.f * gmgmk * gm1sq * a0pq + gmgmk * gmgmk * xz + gm1sq * gm1sq * wy) * rho2;
    float c42 = -(gammk * gamm1 * twgm1 * a0pq + gam * gammk * gammk * xz + gamm1 * gm1sq * wy) * rho;
    float c20 = t * c42;
    float c21 = t * c32;
    float c22 = a0 + 2.f * (cpcq - c00);
    float c23 = t * c12;
    float c24 = t * c02;

    float e0 = e[0], e1 = e[1], e2 = e[2], e3 = e[3], e4 = e[4];
    float ee0 = e0 * c00 + e1 * c10 + e2 * c20 + e3 * c30 + e4 * c40;
    float ee1 = e0 * c01 + e1 * c11 + e2 * c21 + e3 * c31 + e4 * c30;
    float ee2 = e0 * c02 + e1 * c12 + e2 * c22 + e3 * c32 + e4 * c42;
    float ee3 = e0 * c03 + e1 * c13 + e2 * c23 + e3 * c11 + e4 * c10;
    float ee4 = e0 * c04 + e1 * c03 + e2 * c24 + e3 * c01 + e4 * c00;

    float t1 = fmaxf(fmaxf(fmaxf(fabsf(ee0), fabsf(ee1)),
                           fmaxf(fabsf(ee2), fabsf(ee3))), fabsf(ee4));
    t1 = (t1 < 1e-30f) ? 1.f : t1;
    float s = rcpf(t1);
    e[0] = ee0 * s; e[1] = ee1 * s; e[2] = ee2 * s; e[3] = ee3 * s; e[4] = ee4 * s;
}

__device__ __forceinline__ float poly_a(float bb) {
    return 0.9409f + bb * (2.0947f + bb * (-0.8206f + bb * (0.2683f + bb * (-0.0251f))));
}
__device__ __forceinline__ float poly_rho(float aa) {
    return aa * (1.6612f + aa * (-0.4721f + aa * (0.0671f + aa * (-0.0043f + aa * 0.000106f))));
}

// ===========================================================================
// Kernel A: one block per (m,p); compute det over all NC phase velocities and
// write rng(m,p) = max_c det - min_c det.
// ===========================================================================
__global__ void __launch_bounds__(BLK) rng_kernel(
    const float* __restrict__ tlist, const float* __restrict__ d,
    const float* __restrict__ b,     const float* __restrict__ Clist,
    float* __restrict__ rng_out, int P, int L, int NC)
{
    __shared__ float s_rawd[MAXL], s_rawb[MAXL];
    __shared__ float s_xka[MAXL], s_xkb[MAXL], s_gammk[MAXL], s_rho[MAXL], s_dm[MAXL];
    __shared__ float s_mn[BLK / 32], s_mx[BLK / 32];

    const int blk = blockIdx.x;
    const int m = blk / P, p = blk - m * P;
    const int tid = threadIdx.x;

    const float omega_raw = TWO_PI_F / tlist[(size_t)m * P + p];
    const float omega     = fmaxf(omega_raw, 1e-4f);

    // CDNA5 async global->LDS staging of this model's layer tables (ASYNCcnt path)
    if (tid < L) {
        unsigned int ldsd = (unsigned int)(size_t)(&s_rawd[tid]);
        unsigned int ldsb = (unsigned int)(size_t)(&s_rawb[tid]);
        const float* gd = d + (size_t)m * L + tid;
        const float* gb = b + (size_t)m * L + tid;
        asm volatile("global_load_async_to_lds_b32 %0, %1, off" :: "v"(ldsd), "v"(gd) : "memory");
        asm volatile("global_load_async_to_lds_b32 %0, %1, off" :: "v"(ldsb), "v"(gb) : "memory");
        asm volatile("s_wait_asynccnt 0" ::: "memory");
    }
    __syncthreads();

    if (tid < L) {
        float bb = s_rawb[tid];
        float aa = poly_a(bb);
        float tq = bb / omega;
        s_xka[tid]   = omega / aa;
        s_xkb[tid]   = omega / bb;
        s_gammk[tid] = 2.f * tq * tq;
        s_rho[tid]   = poly_rho(aa);
        s_dm[tid]    = s_rawd[tid];
    }
    __syncthreads();

    float mn = __builtin_inff();
    float mx = -__builtin_inff();
    for (int c = tid; c < NC; c += blockDim.x) {
        float wvno  = omega_raw / Clist[c];   // reference builds wvno_g from unclamped omega
        float wvno2 = wvno * wvno;
        float e[5];
        dltar_init(wvno2, s_xka[L - 1], s_xkb[L - 1], s_gammk[L - 1], s_rho[L - 1], e);
        #pragma unroll 1
        for (int l = L - 2; l >= 0; --l)
            dltar_step(wvno, wvno2, s_xka[l], s_xkb[l], s_gammk[l], s_rho[l], s_dm[l], e);
        mn = fminf(mn, e[0]);
        mx = fmaxf(mx, e[0]);
    }

    // wave32 reduction then cross-wave via LDS
    for (int o = 16; o > 0; o >>= 1) {
        mn = fminf(mn, __shfl_xor(mn, o, 32));
        mx = fmaxf(mx, __shfl_xor(mx, o, 32));
    }
    const int lane = tid & 31, wid = tid >> 5;
    if (lane == 0) { s_mn[wid] = mn; s_mx[wid] = mx; }
    __syncthreads();
    if (tid == 0) {
        const int nw = blockDim.x >> 5;
        float amn = s_mn[0], amx = s_mx[0];
        for (int wv = 1; wv < nw; ++wv) { amn = fminf(amn, s_mn[wv]); amx = fmaxf(amx, s_mx[wv]); }
        rng_out[blk] = amx - amn;
    }
}

// ===========================================================================
// Kernel B: one block per m; e00 over P periods + m_v regularizer -> out[m]
// ===========================================================================
__global__ void __launch_bounds__(BLK) final_kernel(
    const float* __restrict__ vlist, const float* __restrict__ tlist,
    const float* __restrict__ d,     const float* __restrict__ b,
    const float* __restrict__ rng_ws, float* __restrict__ out, int P, int L)
{
    __shared__ float s_a[MAXL], s_b[MAXL], s_d[MAXL], s_rho[MAXL];
    __shared__ float s_sum[BLK / 32];

    const int m = blockIdx.x, tid = threadIdx.x;

    if (tid < L) {
        float bb = b[(size_t)m * L + tid];
        float aa = poly_a(bb);
        s_a[tid]   = aa;
        s_b[tid]   = bb;
        s_rho[tid] = poly_rho(aa);
        s_d[tid]   = d[(size_t)m * L + tid];
    }
    __syncthreads();

    float acc = 0.f;
    for (int p = tid; p < P; p += blockDim.x) {
        float omega_raw = TWO_PI_F / tlist[(size_t)m * P + p];
        float omega     = fmaxf(omega_raw, 1e-4f);
        float wvno      = omega_raw / vlist[(size_t)m * P + p];
        float wvno2     = wvno * wvno;
        float e[5];
        {
            float bb = s_b[L - 1];
            float tq = bb / omega;
            dltar_init(wvno2, omega / s_a[L - 1], omega / bb, 2.f * tq * tq, s_rho[L - 1], e);
        }
        #pragma unroll 1
        for (int l = L - 2; l >= 0; --l) {
            float bb = s_b[l];
            float tq = bb / omega;
            dltar_step(wvno, wvno2, omega / s_a[l], omega / bb, 2.f * tq * tq,
                       s_rho[l], s_d[l], e);
        }
        float e00 = e[0] / rng_ws[(size_t)m * P + p];
        float v   = __expf(LN01_F * fabsf(e00)) - 1.f;   // 0.1^|e00| - 1
        acc += fabsf(v);
    }

    for (int o = 16; o > 0; o >>= 1) acc += __shfl_xor(acc, o, 32);
    const int lane = tid & 31, wid = tid >> 5;
    if (lane == 0) s_sum[wid] = acc;
    __syncthreads();
    if (tid == 0) {
        const int nw = blockDim.x >> 5;
        float total = s_sum[0];
        for (int wv = 1; wv < nw; ++wv) total += s_sum[wv];
        float res = total / (float)P;

        // m_v = |Lmat_v @ b[m]| / L   (DAMP_VERTICAL = 1, DAMP_HORIZONTAL = 0)
        float invL = 1.f / (float)L;
        float mv = 0.f;
        for (int i = 0; i < L; ++i) {
            float lv;
            if (i == 0)           lv = s_b[0]     - s_b[1];
            else if (i == L - 1)  lv = s_b[L - 1] - s_b[L - 2];
            else                  lv = 2.f * s_b[i] - s_b[i - 1] - s_b[i + 1];
            mv += fabsf(lv * invL);
        }
        out[m] = res + mv;
    }
}

// ===========================================================================
extern "C" void kernel_launch(void* const* d_in, const int* in_sizes, int n_in,
                              void* d_out, int out_size, void* d_ws, size_t ws_size,
                              hipStream_t stream) {
    const float* vlist = (const float*)d_in[0];
    const float* tlist = (const float*)d_in[1];
    const float* d     = (const float*)d_in[2];
    const float* b     = (const float*)d_in[3];
    const float* Clist = (const float*)d_in[4];
    float* out = (float*)d_out;

    const int M  = out_size;             // 64
    const int P  = in_sizes[0] / M;      // 100
    const int L  = in_sizes[2] / M;      // 32
    const int NC = in_sizes[4];          // 400

    float* rng_ws = (float*)d_ws;        // M*P floats of scratch

    rng_kernel<<<M * P, BLK, 0, stream>>>(tlist, d, b, Clist, rng_ws, P, L, NC);
    final_kernel<<<M, BLK, 0, stream>>>(vlist, tlist, d, b, rng_ws, out, P, L);
}